// SelfAttentionRouter_37220186587188
// MI455X (gfx1250) — compile-verified
//
#include <hip/hip_runtime.h>

// SelfAttentionRouter for MI455X (gfx1250, wave32).
//
// out[b,j] = sum_i x[b,i,j] * ( softmax_i( rowsum(X·X^T)[i] / 16 ) + bias[i,j] )
//
// HBM-bound: 512MB in + 16MB out => ~23us floor at 23.3 TB/s. One block per
// batch streams the 32KB tile into LDS once (float4 coalesced), computes the
// 32x32 Gram via V_WMMA_F32_16X16X4_F32 (3 tiles, symmetry G10=G01^T, one
// wave per tile), wave0 does the 32-way softmax with __shfl_xor, and all 256
// threads run the fused epilogue out of LDS.

typedef __attribute__((ext_vector_type(2))) float v2f;
typedef __attribute__((ext_vector_type(8))) float v8f;

#define CAPS 32
#define FEAT 256
// LDS X tile row stride: 260 % 64 == 4 -> the 16-row-strided WMMA operand
// fetches (K stepping by 4) hit banks {4m+k .. 4m+k+3}, covering all 64 banks.
#define XS_STRIDE 260
// Gram row stride 33 -> conflict-free per-row reduction.
#define GS_STRIDE 33

__global__ __launch_bounds__(256) void selfattn_router_kernel(
    const float* __restrict__ inp,   // [B, 32, 256]
    const float* __restrict__ bias,  // [32, 256]
    float* __restrict__ out) {       // [B, 256]
  __shared__ float Xs[CAPS * XS_STRIDE];  // 33280 B, padded 32x256 tile
  __shared__ float Gs[CAPS * GS_STRIDE];  // 4224 B, padded 32x32 Gram
  __shared__ float Ps[CAPS];              // softmax probabilities

  const int tid = threadIdx.x;
  const size_t b = blockIdx.x;
  const float* xb = inp + b * (size_t)(CAPS * FEAT);

  // ---- Phase 1: stage X[32][256] into padded LDS, 2048 float4s, coalesced.
  {
    const float4* src = (const float4*)xb;
#pragma unroll
    for (int it = 0; it < 8; ++it) {
      int f4 = tid + it * 256;       // float4 index 0..2047
      float4 v = src[f4];
      int row = f4 >> 6;             // 64 float4 per 256-float row
      int col = (f4 & 63) << 2;
      *(float4*)&Xs[row * XS_STRIDE + col] = v;  // 16B aligned (stride%4==0)
    }
  }
  __syncthreads();

  // ---- Phase 2: G = X·X^T with fp32 WMMA. Tiles (0,0),(0,1),(1,1) on
  // waves 0,1,2; G[1,0] recovered by symmetry when dumping tile (0,1).
  // readfirstlane => wave id is provably scalar: uniform branches, no exec
  // juggling around the WMMA region (EXEC stays all-ones as required).
  const int wave = __builtin_amdgcn_readfirstlane(tid >> 5);
  const int lane = tid & 31;
  if (wave < 3) {
    const int tI = (wave == 2) ? 1 : 0;
    const int tJ = (wave == 0) ? 0 : 1;
    const int half = lane >> 4;      // K phase: half 0 -> K,K+1 ; half 1 -> K+2,K+3
    const int sub = lane & 15;       // A: row M ; B: column N (row of X for X^T)
    const float* arow = &Xs[(tI * 16 + sub) * XS_STRIDE + 2 * half];
    const float* brow = &Xs[(tJ * 16 + sub) * XS_STRIDE + 2 * half];
    v8f acc = {};
#pragma unroll 4
    for (int k = 0; k < FEAT; k += 4) {
      v2f av = *(const v2f*)(arow + k);   // ds_load_2addr_b64, conflict-free
      v2f bv = *(const v2f*)(brow + k);
      // (neg_a, A, neg_b, B, c_mod, C, reuse_a, reuse_b); NEG must be 0 for f32.
      acc = __builtin_amdgcn_wmma_f32_16x16x4_f32(
          false, av, false, bv, (short)0, acc, false, false);
    }
    // C/D layout: VGPR v, lane L -> element (M = v + 8*(L/16), N = L%16).
    const int rbase = tI * 16 + 8 * half;
    const int col = tJ * 16 + sub;
#pragma unroll
    for (int v = 0; v < 8; ++v)
      Gs[(rbase + v) * GS_STRIDE + col] = acc[v];
    if (tI != tJ) {                  // uniform branch: mirror G01 -> G10
#pragma unroll
      for (int v = 0; v < 8; ++v)
        Gs[col * GS_STRIDE + (rbase + v)] = acc[v];  // contiguous in v
    }
  }
  __syncthreads();

  // ---- Phase 3: scores = rowsum(G)/16, softmax over 32 capsules (wave 0).
  if (tid < 32) {
    float s = 0.f;
#pragma unroll
    for (int k = 0; k < CAPS; ++k) s += Gs[tid * GS_STRIDE + k];
    s *= 0.0625f;  // 1/sqrt(256)
    float m = s;
#pragma unroll
    for (int off = 16; off > 0; off >>= 1)
      m = fmaxf(m, __shfl_xor(m, off, 32));
    float e = exp2f((s - m) * 1.4426950408889634f);
    float sum = e;
#pragma unroll
    for (int off = 16; off > 0; off >>= 1)
      sum += __shfl_xor(sum, off, 32);
    Ps[tid] = e / sum;
  }
  __syncthreads();

  // ---- Phase 4: out[j] = sum_i Xs[i][j] * (p[i] + bias[i][j]).
  // Xs reads: consecutive lanes -> consecutive banks (conflict-free).
  // Ps reads: broadcast. bias: coalesced, L2-resident (32KB total).
  float acco = 0.f;
#pragma unroll
  for (int i = 0; i < CAPS; ++i) {
    float xv = Xs[i * XS_STRIDE + tid];
    float aw = Ps[i] + bias[i * FEAT + tid];
    acco = fmaf(xv, aw, acco);
  }
  out[b * (size_t)FEAT + tid] = acco;
}

extern "C" void kernel_launch(void* const* d_in, const int* in_sizes, int n_in,
                              void* d_out, int out_size, void* d_ws, size_t ws_size,
                              hipStream_t stream) {
  const float* inp  = (const float*)d_in[0];   // [B,32,256] f32
  const float* bias = (const float*)d_in[1];   // [32,256] f32
  float* out = (float*)d_out;                  // [B,256] f32
  const int B = in_sizes[0] / (CAPS * FEAT);   // 16384
  selfattn_router_kernel<<<B, 256, 0, stream>>>(inp, bias, out);
}